// TopologyAwareGNN_12317966205309
// MI455X (gfx1250) — compile-verified
//
#include <hip/hip_runtime.h>
#include <hip/hip_bf16.h>

typedef __attribute__((ext_vector_type(2))) float v2f;
typedef __attribute__((ext_vector_type(8))) float v8f;

#define HID 128
#define BROW 144  // packed-B row stride in float2 (128 + 16 pad -> dealiases LDS banks)

// ---------------------------------------------------------------- utilities
__global__ void fill_kernel(float* p, float v, int n) {
    int i = blockIdx.x * blockDim.x + threadIdx.x;
    if (i < n) p[i] = v;
}

__global__ void deg_accum_kernel(const int* __restrict__ dst, float* deg, int E) {
    int e = blockIdx.x * blockDim.x + threadIdx.x;
    if (e < E)
        __hip_atomic_fetch_add(&deg[dst[e]], 1.0f, __ATOMIC_RELAXED,
                               __HIP_MEMORY_SCOPE_AGENT);
}

__global__ void deg_finish_kernel(float* dis, float* inv, int n) {
    int i = blockIdx.x * blockDim.x + threadIdx.x;
    if (i < n) {
        float d = dis[i];          // dis currently holds deg
        dis[i] = rsqrtf(d);
        inv[i] = 1.0f / d;
    }
}

// -------------------------------------------------- WMMA f32 GEMM: C = A@B(+bias)
// A: [16*ntiles, K] row-major, B: [K, 128] row-major, C: [rows, 128].
// 256 threads = 8 waves; each wave computes one 16x128 row stripe.
// B is staged in LDS pre-packed in WMMA fragment order:
//   Bs[(g*2+h)*BROW + n] = { B[4g+2h][n], B[4g+2h+1][n] }  (float2)
// so each b fragment is a single aligned ds_load_b64 into an even VGPR pair.
// All A fragments are preloaded into registers (K templated => no spills).
template <int K>
__global__ __launch_bounds__(256) void gemm_wmma_kernel(
    const float* __restrict__ A, const float* __restrict__ B,
    const float* __restrict__ bias, float* __restrict__ C, int ntiles) {
    __shared__ float Bs[2 * (K / 4) * 2 * BROW];  // K=128: 72 KB

    const int tid = threadIdx.x;
    for (int idx = tid; idx < K * HID; idx += 256) {
        const int k = idx >> 7;
        const int n = idx & (HID - 1);
        const int g = k >> 2;
        const int h = (k >> 1) & 1;
        const int p = k & 1;
        Bs[((g * 2 + h) * BROW + n) * 2 + p] = B[idx];
    }
    __syncthreads();

    const int wave = tid >> 5;
    const int tile = blockIdx.x * 8 + wave;
    if (tile >= ntiles) return;          // wave-uniform: EXEC stays all-ones

    const int lane = tid & 31;
    const int half = lane >> 4;          // 0 | 1
    const int m    = lane & 15;          // row within tile (A/C), col mod 16 (B/C)
    const int kh   = half * 2;           // K sub-offset for this lane half

    const float* arow = A + (size_t)(tile * 16 + m) * K;

    // preload all A fragments (contiguous 8B pairs -> global_load_b64, pipelined)
    v2f afrag[K / 4];
#pragma unroll
    for (int s = 0; s < K / 4; ++s)
        afrag[s] = *(const v2f*)(arow + 4 * s + kh);

    const v2f* __restrict__ bsv = ((const v2f*)Bs) + m;

    v8f acc[8] = {};                     // 8 column tiles of 16
#pragma unroll
    for (int s = 0; s < K / 4; ++s) {
        const v2f* brow = bsv + (size_t)(s * 2 + half) * BROW;
#pragma unroll
        for (int n0 = 0; n0 < 8; ++n0) {
            v2f b = brow[n0 * 16];       // single ds_load_b64, fragment-ready
            acc[n0] = __builtin_amdgcn_wmma_f32_16x16x4_f32(
                false, afrag[s], false, b, (short)0, acc[n0], false, false);
        }
    }

    const int mh = half * 8;             // C: lanes 16-31 hold rows M+8
#pragma unroll
    for (int n0 = 0; n0 < 8; ++n0) {
        const int col = n0 * 16 + m;
        const float bv = bias ? bias[col] : 0.0f;
#pragma unroll
        for (int r = 0; r < 8; ++r) {
            C[(size_t)(tile * 16 + mh + r) * HID + col] = acc[n0][r] + bv;
        }
    }
}

// ---------------------------------------------------- edge gather/scatter-add
// one wave per edge; lane handles 4 consecutive features (float4 = 512B/row)
__global__ __launch_bounds__(256) void scatter_kernel(
    const int* __restrict__ src, const int* __restrict__ dst,
    const float* __restrict__ dis, const float* __restrict__ hp,
    float* __restrict__ agg, int E) {
    const int wave = (int)((blockIdx.x * blockDim.x + threadIdx.x) >> 5);
    const int lane = threadIdx.x & 31;
    if (wave >= E) return;
    const int s = src[wave];
    const int d = dst[wave];
    const float coef = dis[s] * dis[d];
    const float4 v = ((const float4*)(hp + (size_t)s * HID))[lane];
    float* ad = agg + (size_t)d * HID + lane * 4;
    __hip_atomic_fetch_add(ad + 0, v.x * coef, __ATOMIC_RELAXED, __HIP_MEMORY_SCOPE_AGENT);
    __hip_atomic_fetch_add(ad + 1, v.y * coef, __ATOMIC_RELAXED, __HIP_MEMORY_SCOPE_AGENT);
    __hip_atomic_fetch_add(ad + 2, v.z * coef, __ATOMIC_RELAXED, __HIP_MEMORY_SCOPE_AGENT);
    __hip_atomic_fetch_add(ad + 3, v.w * coef, __ATOMIC_RELAXED, __HIP_MEMORY_SCOPE_AGENT);
}

// h = relu(agg + hp*inv[node] + bias[feat])
__global__ void finish_kernel(const float* __restrict__ agg,
                              const float* __restrict__ hp,
                              const float* __restrict__ inv,
                              const float* __restrict__ bias,
                              float* __restrict__ h, int total) {
    int idx = blockIdx.x * blockDim.x + threadIdx.x;
    if (idx >= total) return;
    const int node = idx >> 7;
    const int f    = idx & (HID - 1);
    float v = agg[idx] + hp[idx] * inv[node] + bias[f];
    h[idx] = fmaxf(v, 0.0f);
}

// column sums of h into gsum[128]
__global__ __launch_bounds__(128) void mean_accum_kernel(
    const float* __restrict__ h, float* gsum, int n) {
    const int f = threadIdx.x;
    float s = 0.0f;
    for (int r = blockIdx.x; r < n; r += gridDim.x) s += h[(size_t)r * HID + f];
    __hip_atomic_fetch_add(&gsum[f], s, __ATOMIC_RELAXED, __HIP_MEMORY_SCOPE_AGENT);
}

// g = gsum/N; t = relu(g@fc1+b1); out = t@fc2+b2
__global__ __launch_bounds__(128) void head_kernel(
    const float* __restrict__ gsum,
    const float* __restrict__ fc1w, const float* __restrict__ fc1b,
    const float* __restrict__ fc2w, const float* __restrict__ fc2b,
    float* __restrict__ out, float invN) {
    __shared__ float g[HID];
    __shared__ float t[HID];
    const int tid = threadIdx.x;
    g[tid] = gsum[tid] * invN;
    __syncthreads();
    float s = fc1b[tid];
    for (int i = 0; i < HID; ++i) s += g[i] * fc1w[i * HID + tid];
    t[tid] = fmaxf(s, 0.0f);
    __syncthreads();
    if (tid < 64) {
        float o = fc2b[tid];
        for (int j = 0; j < HID; ++j) o += t[j] * fc2w[j * 64 + tid];
        out[tid] = o;
    }
}

// ------------------------------------------------------------------- driver
extern "C" void kernel_launch(void* const* d_in, const int* in_sizes, int n_in,
                              void* d_out, int out_size, void* d_ws, size_t ws_size,
                              hipStream_t stream) {
    const float* x      = (const float*)d_in[0];
    const int*   ei     = (const int*)d_in[1];
    const float* emb_w  = (const float*)d_in[2];
    const float* emb_b  = (const float*)d_in[3];
    const float* conv_w = (const float*)d_in[4];
    const float* conv_b = (const float*)d_in[5];
    const float* fc1_w  = (const float*)d_in[6];
    const float* fc1_b  = (const float*)d_in[7];
    const float* fc2_w  = (const float*)d_in[8];
    const float* fc2_b  = (const float*)d_in[9];

    const int N = in_sizes[0] / 16;       // 100000
    const int E = in_sizes[1] / 2;        // 1600000
    const int* src = ei;
    const int* dst = ei + E;

    float* wsf  = (float*)d_ws;
    float* dis  = wsf;                                  // [N] deg -> deg^-1/2
    float* inv  = wsf + N;                              // [N] 1/deg
    float* h    = wsf + 2 * (size_t)N;                  // [N,128]
    float* hp   = h  + (size_t)N * HID;                 // [N,128]
    float* agg  = hp + (size_t)N * HID;                 // [N,128]
    float* gsum = agg + (size_t)N * HID;                // [128]

    const int ntiles  = (N + 15) / 16;                  // N%16==0 here
    const int gblocks = (ntiles + 7) / 8;
    const int totalNH = N * HID;

    // degrees (dst-only, + self loop)
    fill_kernel<<<(N + 255) / 256, 256, 0, stream>>>(dis, 1.0f, N);
    deg_accum_kernel<<<(E + 255) / 256, 256, 0, stream>>>(dst, dis, E);
    deg_finish_kernel<<<(N + 255) / 256, 256, 0, stream>>>(dis, inv, N);

    // node embedding: h = x @ emb_w + emb_b   (K=16)
    gemm_wmma_kernel<16><<<gblocks, 256, 0, stream>>>(x, emb_w, emb_b, h, ntiles);

    // 3 GCN layers
    for (int l = 0; l < 3; ++l) {
        gemm_wmma_kernel<128><<<gblocks, 256, 0, stream>>>(
            h, conv_w + (size_t)l * HID * HID, nullptr, hp, ntiles);
        fill_kernel<<<(totalNH + 255) / 256, 256, 0, stream>>>(agg, 0.0f, totalNH);
        scatter_kernel<<<(E + 7) / 8, 256, 0, stream>>>(src, dst, dis, hp, agg, E);
        finish_kernel<<<(totalNH + 255) / 256, 256, 0, stream>>>(
            agg, hp, inv, conv_b + l * HID, h, totalNH);
    }

    // global mean + MLP head
    fill_kernel<<<1, 128, 0, stream>>>(gsum, 0.0f, HID);
    mean_accum_kernel<<<512, 128, 0, stream>>>(h, gsum, N);
    head_kernel<<<1, 128, 0, stream>>>(gsum, fc1_w, fc1_b, fc2_w, fc2_b,
                                       (float*)d_out, 1.0f / (float)N);
}